// SAGE_16209206575323
// MI455X (gfx1250) — compile-verified
//
#include <hip/hip_runtime.h>

#define DI 128   // channels

typedef __attribute__((ext_vector_type(2))) float v2f;
typedef __attribute__((ext_vector_type(8))) float v8f;

__device__ __forceinline__ void atomAddF(float* p, float v) {
    __hip_atomic_fetch_add(p, v, __ATOMIC_RELAXED, __HIP_MEMORY_SCOPE_AGENT);
}

// ---- degree count: one thread per edge -------------------------------------
__global__ __launch_bounds__(256) void degree_kernel(const int* __restrict__ dst,
                                                     float* __restrict__ cnt, int E) {
    int e = blockIdx.x * blockDim.x + threadIdx.x;
    if (e < E) atomAddF(cnt + dst[e], 1.0f);
}

// ---- cnt -> 1/max(cnt,1) in place ------------------------------------------
__global__ __launch_bounds__(256) void recip_kernel(float* __restrict__ cnt, int n) {
    int i = blockIdx.x * blockDim.x + threadIdx.x;
    if (i < n) cnt[i] = 1.0f / fmaxf(cnt[i], 1.0f);
}

// ---- edge scatter-add: one wave32 per edge, lane handles 4 channels --------
__global__ __launch_bounds__(256) void scatter_kernel(const float* __restrict__ h,
                                                      const int* __restrict__ src,
                                                      const int* __restrict__ dst,
                                                      float* __restrict__ agg, int E) {
    int e = blockIdx.x * (blockDim.x >> 5) + (threadIdx.x >> 5);
    if (e >= E) return;
    int lane = threadIdx.x & 31;
    int s = src[e], d = dst[e];
    const float4 v = *(const float4*)(h + (size_t)s * DI + lane * 4);
    float* p = agg + (size_t)d * DI + lane * 4;
    atomAddF(p + 0, v.x);
    atomAddF(p + 1, v.y);
    atomAddF(p + 2, v.z);
    atomAddF(p + 3, v.w);
}

// ---- fused SAGE linear: out = (agg*rdeg)@Wl + h@Wr + bl --------------------
// Block = 256 threads = 8 waves; 64 node rows per block (4 row-tiles of 16).
// Wave w owns output cols [16w,16w+16) for all 4 row tiles -> 4 independent
// accumulators reuse each B fragment 4x. A (64x256 fp32, 64 KB) staged in LDS.
__global__ __launch_bounds__(256) void sage_gemm_kernel(
        const float* __restrict__ agg, const float* __restrict__ h,
        const float* __restrict__ rdeg,
        const float* __restrict__ Wl, const float* __restrict__ Wr,
        const float* __restrict__ bl,
        float* __restrict__ o0, float* __restrict__ o1,
        int relu0, int relu1, int Nn) {
    __shared__ float sA[64 * 256];   // 64 KB: cols 0..127 = agg*rdeg, 128..255 = h

    int tid  = threadIdx.x;
    int row0 = blockIdx.x * 64;

    // stage A tile (coalesced in k)
    for (int idx = tid; idx < 64 * 256; idx += 256) {
        int r = idx >> 8;
        int k = idx & 255;
        int gm = row0 + r;
        float v = 0.0f;
        if (gm < Nn) {
            if (k < DI) v = agg[(size_t)gm * DI + k] * rdeg[gm];
            else        v = h[(size_t)gm * DI + (k - DI)];
        }
        sA[idx] = v;
    }
    __syncthreads();

    int lane  = tid & 31;
    int wave  = tid >> 5;
    int ln15  = lane & 15;              // A row (m) / B col (n) within tile
    int khalf = (lane >> 4) << 1;       // 0 or 2 : K offset held by this lane
    int colb  = wave * 16;

    v8f acc0 = {}, acc1 = {}, acc2 = {}, acc3 = {};

    // agg part: A cols 0..127 vs Wl
#pragma unroll
    for (int ks = 0; ks < 32; ++ks) {
        int k = ks * 4 + khalf;
        v2f b;
        b.x = Wl[(size_t)k * DI + colb + ln15];
        b.y = Wl[(size_t)(k + 1) * DI + colb + ln15];
        v2f a0, a1, a2, a3;
        a0.x = sA[(0 * 16 + ln15) * 256 + k];  a0.y = sA[(0 * 16 + ln15) * 256 + k + 1];
        a1.x = sA[(1 * 16 + ln15) * 256 + k];  a1.y = sA[(1 * 16 + ln15) * 256 + k + 1];
        a2.x = sA[(2 * 16 + ln15) * 256 + k];  a2.y = sA[(2 * 16 + ln15) * 256 + k + 1];
        a3.x = sA[(3 * 16 + ln15) * 256 + k];  a3.y = sA[(3 * 16 + ln15) * 256 + k + 1];
        acc0 = __builtin_amdgcn_wmma_f32_16x16x4_f32(false, a0, false, b, (short)0, acc0, false, false);
        acc1 = __builtin_amdgcn_wmma_f32_16x16x4_f32(false, a1, false, b, (short)0, acc1, false, false);
        acc2 = __builtin_amdgcn_wmma_f32_16x16x4_f32(false, a2, false, b, (short)0, acc2, false, false);
        acc3 = __builtin_amdgcn_wmma_f32_16x16x4_f32(false, a3, false, b, (short)0, acc3, false, false);
    }
    // h part: A cols 128..255 vs Wr
#pragma unroll
    for (int ks = 0; ks < 32; ++ks) {
        int k = ks * 4 + khalf;
        v2f b;
        b.x = Wr[(size_t)k * DI + colb + ln15];
        b.y = Wr[(size_t)(k + 1) * DI + colb + ln15];
        v2f a0, a1, a2, a3;
        a0.x = sA[(0 * 16 + ln15) * 256 + 128 + k];  a0.y = sA[(0 * 16 + ln15) * 256 + 128 + k + 1];
        a1.x = sA[(1 * 16 + ln15) * 256 + 128 + k];  a1.y = sA[(1 * 16 + ln15) * 256 + 128 + k + 1];
        a2.x = sA[(2 * 16 + ln15) * 256 + 128 + k];  a2.y = sA[(2 * 16 + ln15) * 256 + 128 + k + 1];
        a3.x = sA[(3 * 16 + ln15) * 256 + 128 + k];  a3.y = sA[(3 * 16 + ln15) * 256 + 128 + k + 1];
        acc0 = __builtin_amdgcn_wmma_f32_16x16x4_f32(false, a0, false, b, (short)0, acc0, false, false);
        acc1 = __builtin_amdgcn_wmma_f32_16x16x4_f32(false, a1, false, b, (short)0, acc1, false, false);
        acc2 = __builtin_amdgcn_wmma_f32_16x16x4_f32(false, a2, false, b, (short)0, acc2, false, false);
        acc3 = __builtin_amdgcn_wmma_f32_16x16x4_f32(false, a3, false, b, (short)0, acc3, false, false);
    }

    float bias  = bl[colb + ln15];
    int   mbase = (lane >> 4) * 8;      // C/D layout: VGPR r -> row r (+8 for hi lanes)
    v8f accs[4] = {acc0, acc1, acc2, acc3};
#pragma unroll
    for (int t = 0; t < 4; ++t) {
#pragma unroll
        for (int r = 0; r < 8; ++r) {
            int gm = row0 + t * 16 + mbase + r;
            if (gm < Nn) {
                float v = accs[t][r] + bias;
                size_t off = (size_t)gm * DI + colb + ln15;
                if (o0) o0[off] = relu0 ? fmaxf(v, 0.0f) : v;
                if (o1) o1[off] = relu1 ? fmaxf(v, 0.0f) : v;
            }
        }
    }
}

// ---- cluster pooling: wave per node, atomic add row into g[cluster] --------
__global__ __launch_bounds__(256) void pool_scatter_kernel(const float* __restrict__ h,
                                                           const int* __restrict__ cluster,
                                                           float* __restrict__ g,
                                                           float* __restrict__ ccnt, int Nn) {
    int i = blockIdx.x * (blockDim.x >> 5) + (threadIdx.x >> 5);
    if (i >= Nn) return;
    int lane = threadIdx.x & 31;
    int c = cluster[i];
    const float4 v = *(const float4*)(h + (size_t)i * DI + lane * 4);
    float* p = g + (size_t)c * DI + lane * 4;
    atomAddF(p + 0, v.x);
    atomAddF(p + 1, v.y);
    atomAddF(p + 2, v.z);
    atomAddF(p + 3, v.w);
    if (lane == 0) atomAddF(ccnt + c, 1.0f);
}

__global__ __launch_bounds__(256) void pool_scale_kernel(float* __restrict__ g,
                                                         const float* __restrict__ ccnt, int C) {
    int idx = blockIdx.x * blockDim.x + threadIdx.x;
    if (idx < C * DI) g[idx] *= 1.0f / fmaxf(ccnt[idx >> 7], 1.0f);
}

extern "C" void kernel_launch(void* const* d_in, const int* in_sizes, int n_in,
                              void* d_out, int out_size, void* d_ws, size_t ws_size,
                              hipStream_t stream) {
    const float* x       = (const float*)d_in[0];
    const int*   ei      = (const int*)d_in[1];
    const int*   cluster = (const int*)d_in[2];
    const float* Wl0 = (const float*)d_in[3];
    const float* bl0 = (const float*)d_in[4];
    const float* Wr0 = (const float*)d_in[5];
    const float* Wl1 = (const float*)d_in[6];
    const float* bl1 = (const float*)d_in[7];
    const float* Wr1 = (const float*)d_in[8];
    const float* Wl2 = (const float*)d_in[9];
    const float* bl2 = (const float*)d_in[10];
    const float* Wr2 = (const float*)d_in[11];

    const int N = in_sizes[0] / DI;
    const int E = in_sizes[1] / 2;
    const int C = (out_size - 3 * N * DI) / DI;

    const int* src = ei;
    const int* dst = ei + E;

    float* xf0  = (float*)d_out;                        // slot0: h1, then final xf
    float* out1 = (float*)d_out + (size_t)N * DI;       // slot1: pre-ReLU layer-1 out
    float* gbuf = (float*)d_out + (size_t)2 * N * DI;   // slot2: pooled g [C,D]
    float* xf1  = gbuf + (size_t)C * DI;                // slot3: h2, then xf copy

    float* agg  = (float*)d_ws;                         // [N, D]
    float* rdeg = agg + (size_t)N * DI;                 // [N]
    float* ccnt = rdeg + N;                             // [C]

    const int tpb = 256;
    dim3 blk(tpb);
    int edgeWaveGrid = (E + (tpb / 32) - 1) / (tpb / 32);
    int nodeWaveGrid = (N + (tpb / 32) - 1) / (tpb / 32);
    int gemmGrid     = (N + 63) / 64;

    // in-degree reciprocals (once)
    hipMemsetAsync(rdeg, 0, (size_t)N * sizeof(float), stream);
    degree_kernel<<<(E + tpb - 1) / tpb, blk, 0, stream>>>(dst, rdeg, E);
    recip_kernel<<<(N + tpb - 1) / tpb, blk, 0, stream>>>(rdeg, N);

    // ---- layer 0: h1 = relu(conv0(x)) -> slot0
    hipMemsetAsync(agg, 0, (size_t)N * DI * sizeof(float), stream);
    scatter_kernel<<<edgeWaveGrid, blk, 0, stream>>>(x, src, dst, agg, E);
    sage_gemm_kernel<<<gemmGrid, blk, 0, stream>>>(agg, x, rdeg, Wl0, Wr0, bl0,
                                                   xf0, nullptr, 1, 0, N);

    // ---- layer 1: out1 = conv1(h1) -> slot1 (raw); h2 = relu -> slot3
    hipMemsetAsync(agg, 0, (size_t)N * DI * sizeof(float), stream);
    scatter_kernel<<<edgeWaveGrid, blk, 0, stream>>>(xf0, src, dst, agg, E);
    sage_gemm_kernel<<<gemmGrid, blk, 0, stream>>>(agg, xf0, rdeg, Wl1, Wr1, bl1,
                                                   out1, xf1, 0, 1, N);

    // ---- cluster mean pooling of h2 -> slot2
    hipMemsetAsync(gbuf, 0, (size_t)C * DI * sizeof(float), stream);
    hipMemsetAsync(ccnt, 0, (size_t)C * sizeof(float), stream);
    pool_scatter_kernel<<<nodeWaveGrid, blk, 0, stream>>>(xf1, cluster, gbuf, ccnt, N);
    pool_scale_kernel<<<(C * DI + tpb - 1) / tpb, blk, 0, stream>>>(gbuf, ccnt, C);

    // ---- layer 2: xf = conv2(h2) -> slot0 and slot3 (in-place over h2; each
    // block stages its h rows to LDS before storing, rows are block-private)
    hipMemsetAsync(agg, 0, (size_t)N * DI * sizeof(float), stream);
    scatter_kernel<<<edgeWaveGrid, blk, 0, stream>>>(xf1, src, dst, agg, E);
    sage_gemm_kernel<<<gemmGrid, blk, 0, stream>>>(agg, xf1, rdeg, Wl2, Wr2, bl2,
                                                   xf0, xf1, 0, 0, N);
}